// BagRE_75900662055642
// MI455X (gfx1250) — compile-verified
//
#include <hip/hip_runtime.h>
#include <hip/hip_bf16.h>

// BagRE: segment-mean over sorted bags + linear classifier.
// Stage 1: bandwidth-bound coalesced segment mean (256 MB stream).
// Stage 2: fp32 WMMA GEMM  out[g,c] = sum_h mean[g,h]*W[c,h] + b[c].

#define HDIM 256
#define CDIM 128
#define GBAGS 8192

typedef __attribute__((ext_vector_type(2))) float v2f;
typedef __attribute__((ext_vector_type(8))) float v8f;

// ---------------- Stage 1: segment mean (one block per bag) ----------------
__global__ __launch_bounds__(HDIM) void bagre_seg_mean(
    const float* __restrict__ hidden,
    const int*   __restrict__ bag_id,
    float*       __restrict__ mean,
    int n_rows)
{
    const int g = blockIdx.x;      // bag id
    const int h = threadIdx.x;     // column 0..255

    // lower_bound(g): bag_id sorted ascending. Uniform across block -> the
    // loads broadcast from one address; ~18 iterations.
    int lo = 0, hi = n_rows;
    while (lo < hi) {
        int mid = (lo + hi) >> 1;
        if (bag_id[mid] < g) lo = mid + 1; else hi = mid;
    }
    const int start = lo;
    // lower_bound(g+1)
    hi = n_rows;
    while (lo < hi) {
        int mid = (lo + hi) >> 1;
        if (bag_id[mid] < g + 1) lo = mid + 1; else hi = mid;
    }
    const int end = lo;

    float sum = 0.0f;
    // 256 threads read a contiguous 1 KB line per row: fully coalesced.
    for (int r = start; r < end; ++r)
        sum += hidden[r * HDIM + h];

    const float cnt = (float)(end - start);
    mean[g * HDIM + h] = sum / fmaxf(cnt, 1.0f);
}

// ---------------- Stage 2: WMMA f32 GEMM, one wave per 16x16 tile ----------
// grid.x = GBAGS/16, grid.y = CDIM/16, block = 32 (one wave, EXEC all 1s).
__global__ __launch_bounds__(32) void bagre_gemm_wmma(
    const float* __restrict__ mean,   // [GBAGS, HDIM]
    const float* __restrict__ W,      // [CDIM, HDIM]
    const float* __restrict__ bias,   // [CDIM]
    float*       __restrict__ out)    // [GBAGS, CDIM]
{
    const int gt   = blockIdx.x << 4;          // bag-tile base (M)
    const int ct   = blockIdx.y << 4;          // class-tile base (N)
    const int lane = threadIdx.x;              // 0..31
    const int half = lane >> 4;                // K sub-select
    const int l16  = lane & 15;

    // A lane row: mean[gt+l16, :]   (A is 16x4 per step; lanes0-15 K=k0..k0+1,
    //                                lanes16-31 K=k0+2..k0+3)
    // B lane row: W[ct+l16, :]      (B = W^T, mirrored layout -> same indexing)
    const float* __restrict__ arow = mean + (size_t)(gt + l16) * HDIM;
    const float* __restrict__ brow = W    + (size_t)(ct + l16) * HDIM;

    v8f acc = {};
#pragma unroll 8
    for (int k0 = 0; k0 < HDIM; k0 += 4) {
        const int kk = k0 + (half << 1);               // even -> 8B aligned
        const v2f a = *(const v2f*)(arow + kk);
        const v2f b = *(const v2f*)(brow + kk);
        // 8 args: (neg_a, A, neg_b, B, c_mod, C, reuse_a, reuse_b)
        acc = __builtin_amdgcn_wmma_f32_16x16x4_f32(
            /*neg_a=*/false, a, /*neg_b=*/false, b,
            /*c_mod=*/(short)0, acc, /*reuse_a=*/false, /*reuse_b=*/false);
    }

    const float bv = bias[ct + l16];                   // N = l16 for this lane
#pragma unroll
    for (int v = 0; v < 8; ++v) {
        const int m = v + (half << 3);                 // M = v + 8*half
        out[(size_t)(gt + m) * CDIM + ct + l16] = acc[v] + bv;
    }
}

extern "C" void kernel_launch(void* const* d_in, const int* in_sizes, int n_in,
                              void* d_out, int out_size, void* d_ws, size_t ws_size,
                              hipStream_t stream) {
    const float* hidden = (const float*)d_in[0];   // [N, 256] f32
    const float* W      = (const float*)d_in[1];   // [128, 256] f32
    const float* b      = (const float*)d_in[2];   // [128] f32
    const int*   bag_id = (const int*)d_in[3];     // [N] sorted
    float*       mean   = (float*)d_ws;            // [8192, 256] f32 = 8 MB scratch
    float*       out    = (float*)d_out;           // [8192, 128] f32

    const int n_rows = in_sizes[0] / HDIM;

    bagre_seg_mean<<<GBAGS, HDIM, 0, stream>>>(hidden, bag_id, mean, n_rows);

    dim3 grid(GBAGS / 16, CDIM / 16);
    bagre_gemm_wmma<<<grid, 32, 0, stream>>>(mean, W, b, out);
}